// DualChannelLearner_44306882625856
// MI455X (gfx1250) — compile-verified
//
#include <hip/hip_runtime.h>
#include <hip/hip_bf16.h>

// ---------------------------------------------------------------------------
// DualChannelLearner for MI455X (gfx1250, wave32).
// Memory-bound (~3 FLOP/byte, ~1.3 GB total traffic -> ~60us floor at
// 23.3 TB/s): keep fp32 end-to-end, use V_WMMA_F32_16X16X4_F32 for the two
// GEMM-shaped contractions (f1@w, K=307; temporal mixing, K=T<=13).
// All WMMA operand loads are unconditional (padded buffers / clamped index +
// cndmask) so the inner loops carry no exec-mask branching.
// ---------------------------------------------------------------------------

#define B_    32
#define C_    128
#define N_    307
#define KPAD_ 308               // N_ padded to even for b64 A-loads / no guards
#define L_    12
#define LH_   13
#define CN_   (C_ * N_)         // 39296 rows per batch

typedef __attribute__((ext_vector_type(2))) float v2f;
typedef __attribute__((ext_vector_type(8))) float v8f;

// ---- f1[b,t,n] = sum_c seq[b,c,n,t] * c1[c]  (row stride KPAD_, zero tail) -
template<int T>
__global__ void k_f1(const float* __restrict__ seq, const float* __restrict__ c1,
                     float* __restrict__ f1) {
    int b   = blockIdx.y;
    int idx = blockIdx.x * blockDim.x + threadIdx.x;   // idx over n*T + t
    if (blockIdx.x == 0 && idx < T)                    // zero the k=307 pad col
        f1[((size_t)b * T + idx) * KPAD_ + N_] = 0.f;
    if (idx >= N_ * T) return;
    int n = idx / T, t = idx % T;
    const float* base = seq + (size_t)b * C_ * (size_t)(N_ * T);
    float acc = 0.f;
#pragma unroll 4
    for (int c = 0; c < C_; ++c)
        acc = fmaf(base[(size_t)c * (N_ * T) + idx], c1[c], acc);
    f1[((size_t)b * T + t) * KPAD_ + n] = acc;
}

// ---- f2[b,t,c] = sum_n seq[b,c,n,t] * c2[n] --------------------------------
// block = (32 * T) threads, tid = w*T + t (t fastest) => fully coalesced reads.
template<int T>
__global__ void k_f2(const float* __restrict__ seq, const float* __restrict__ c2,
                     float* __restrict__ f2) {
    __shared__ float red[32 * T];
    int b = blockIdx.y, c = blockIdx.x;
    int tid = threadIdx.x;
    int t = tid % T, w = tid / T;                      // w in [0,32)
    const float* base = seq + (((size_t)b * C_ + c) * (size_t)N_) * T;
    float acc = 0.f;
    for (int n = w; n < N_; n += 32)
        acc = fmaf(base[n * T + t], c2[n], acc);
    red[tid] = acc;
    __syncthreads();
    if (tid < T) {
        float s = 0.f;
#pragma unroll
        for (int ww = 0; ww < 32; ++ww) s += red[ww * T + tid];
        f2[((size_t)b * T + tid) * C_ + c] = s;
    }
}

// ---- zero-padded copy of w:  wpad[308][128], rows >= 307 are zero ----------
__global__ void k_wcopy(const float* __restrict__ wmat, float* __restrict__ wpad) {
    int idx = blockIdx.x * 256 + threadIdx.x;          // over KPAD_*C_
    if (idx >= KPAD_ * C_) return;
    int n = idx / C_;
    wpad[idx] = (n < N_) ? wmat[idx] : 0.f;
}

// ---- f1w[row, c] = sum_n f1[row, n] * w[n, c] ; row = b*T+t ----------------
// GEMM M=B*T (384/416, /16 exact), N=C=128, K=308 (padded). One wave / tile.
// No conditionals in the K-loop; A is an aligned b64 load.
template<int T>
__global__ void k_f1w(const float* __restrict__ f1, const float* __restrict__ wpad,
                      float* __restrict__ f1w) {
    int lane    = threadIdx.x;          // 0..31
    int hi      = lane >> 4;            // lane group
    int m       = lane & 15;            // M (for A) / N (for B,D)
    int rowbase = blockIdx.x * 16;
    int cbase   = blockIdx.y * 16;
    const float* arow = f1 + (size_t)(rowbase + m) * KPAD_;
    const float* bcol = wpad + cbase + m;
    v8f acc = {};
#pragma unroll 4
    for (int k0 = 0; k0 < KPAD_; k0 += 4) {
        int ka = k0 + 2 * hi;           // ISA 16x4 f32 A layout: v0->K=ka, v1->K=ka+1
        v2f a = *(const v2f*)(arow + ka);          // aligned: KPAD_ even, ka even
        v2f bv;
        bv.x = bcol[(size_t)ka * C_];
        bv.y = bcol[(size_t)(ka + 1) * C_];
        acc = __builtin_amdgcn_wmma_f32_16x16x4_f32(false, a, false, bv,
                                                    (short)0, acc, false, false);
    }
#pragma unroll
    for (int i = 0; i < 8; ++i) {       // D layout: VGPR i, M = i + 8*hi, N = m
        int mm = i + 8 * hi;
        f1w[(size_t)(rowbase + mm) * C_ + cbase + m] = acc[i];
    }
}

// ---- scores -> logits -> BatchNorm(b,t stats) -> softmax -> cpad -----------
// Tiny (<= 5408 elements); one block so BN's global (b,t) statistics stay
// local.  Emits a zero-padded 16x16 coefs tile per batch: cpad[b][q][t].
template<int T>
__global__ void k_attn(const float* __restrict__ f1w, const float* __restrict__ f2,
                       const float* __restrict__ bmat, const float* __restrict__ vmat,
                       const float* __restrict__ gamma, const float* __restrict__ beta,
                       float* __restrict__ cpad) {
    __shared__ float sc[B_ * T * T];
    __shared__ float lg[B_ * T * T];
    __shared__ float scaleS[T], shiftS[T];
    const int tid = threadIdx.x, nth = blockDim.x;
    const int TOT = B_ * T * T;

    for (int idx = tid; idx < B_ * 256; idx += nth) cpad[idx] = 0.f;  // pad zeros

    for (int idx = tid; idx < TOT; idx += nth) {
        int b = idx / (T * T), r = idx % (T * T), t = r / T, q = r % T;
        const float* fa = f1w + ((size_t)b * T + t) * C_;
        const float* fb = f2  + ((size_t)b * T + q) * C_;
        float acc = bmat[t * T + q];
#pragma unroll 4
        for (int c = 0; c < C_; ++c) acc = fmaf(fa[c], fb[c], acc);
        sc[idx] = 1.f / (1.f + __expf(-acc));          // sigmoid
    }
    __syncthreads();
    for (int idx = tid; idx < TOT; idx += nth) {       // logits = v @ scores
        int b = idx / (T * T), r = idx % (T * T), t = r / T, q = r % T;
        float acc = 0.f;
#pragma unroll
        for (int k = 0; k < T; ++k) acc = fmaf(vmat[t * T + k], sc[b * T * T + k * T + q], acc);
        lg[idx] = acc;
    }
    __syncthreads();
    if (tid < T) {                                     // BN over (b,t) per q
        float mean = 0.f, m2 = 0.f;
        for (int i = 0; i < B_ * T; ++i) { float x = lg[i * T + tid]; mean += x; m2 += x * x; }
        const float inv = 1.f / (float)(B_ * T);
        mean *= inv;
        float var = m2 * inv - mean * mean;
        float s = gamma[tid] * rsqrtf(var + 1e-5f);
        scaleS[tid] = s;
        shiftS[tid] = beta[tid] - mean * s;
    }
    __syncthreads();
    for (int row = tid; row < B_ * T; row += nth) {    // softmax over q
        int b = row / T, t = row % T;
        float mx = -3.0e38f;
#pragma unroll
        for (int q = 0; q < T; ++q) mx = fmaxf(mx, lg[row * T + q] * scaleS[q] + shiftS[q]);
        float sum = 0.f;
#pragma unroll
        for (int q = 0; q < T; ++q) sum += __expf(lg[row * T + q] * scaleS[q] + shiftS[q] - mx);
        float rs = 1.f / sum;
#pragma unroll
        for (int q = 0; q < T; ++q)                    // transposed, padded store
            cpad[b * 256 + q * 16 + t] =
                __expf(lg[row * T + q] * scaleS[q] + shiftS[q] - mx) * rs;
    }
}

// ---- out[b,r,q] = sum_t seq[b,r,t] * coefs[b,q,t]  (dominant, memory-bound) -
// Per batch: (C*N=39296) x T times T x T. 8 waves/block, one 16-row tile each;
// 2456 tiles = 307 blocks * 8 waves (exact).  B comes from the zero-padded
// cpad tile (unconditional aligned b64); A tail (T=13) uses clamped-index
// loads + cndmask zeroing -> no exec-mask branches anywhere in the loop.
template<int T>
__global__ void k_mix(const float* __restrict__ seq, const float* __restrict__ cpad,
                      float* __restrict__ out) {
    int lane = threadIdx.x & 31;
    int wave = threadIdx.x >> 5;
    int b    = blockIdx.y;
    int tile = blockIdx.x * 8 + wave;
    size_t rowbase = (size_t)b * CN_ + (size_t)tile * 16;
    int hi = lane >> 4, m = lane & 15;
    const float* cb   = cpad + (size_t)b * 256;        // [q][t], 16x16, zero-padded
    const float* arow = seq + (rowbase + m) * T;
    v8f acc = {};
    constexpr int KSTEPS = (T + 3) / 4;
#pragma unroll
    for (int s = 0; s < KSTEPS; ++s) {
        int ka = s * 4 + 2 * hi;
        v2f bv = *(const v2f*)(cb + m * 16 + ka);      // aligned, unconditional
        v2f a;
        if (4 * (s + 1) <= T) {                        // compile-time per step
            if constexpr ((T & 1) == 0) {
                a = *(const v2f*)(arow + ka);          // aligned: T even, ka even
            } else {
                a.x = arow[ka];
                a.y = arow[ka + 1];
            }
        } else {                                       // tail: clamp + cndmask
            int k0c = (ka     < T - 1) ? ka       : (T - 1);
            int k1c = (ka + 1 < T - 1) ? (ka + 1) : (T - 1);
            float x0 = arow[k0c], x1 = arow[k1c];
            a.x = (ka     < T) ? x0 : 0.f;
            a.y = (ka + 1 < T) ? x1 : 0.f;
        }
        acc = __builtin_amdgcn_wmma_f32_16x16x4_f32(false, a, false, bv,
                                                    (short)0, acc, false, false);
    }
    if (m < T) {                                       // q = m
#pragma unroll
        for (int i = 0; i < 8; ++i) {
            int mm = i + 8 * hi;
            out[(rowbase + mm) * T + m] = acc[i];
        }
    }
}

// ---- utility kernels -------------------------------------------------------
__global__ void k_pad(const float* __restrict__ XH, float* __restrict__ outp) {
    size_t idx = (size_t)blockIdx.x * 256 + threadIdx.x;
    if (idx >= (size_t)B_ * CN_ * LH_) return;
    int t = (int)(idx % LH_);
    size_t row = idx / LH_;
    outp[idx] = (t == 0) ? 0.f : XH[row * L_ + (t - 1)];
}

__global__ void k_reslast(const float* __restrict__ XL, const float* __restrict__ seqfin,
                          float* __restrict__ lowlast) {
    size_t i = (size_t)blockIdx.x * 256 + threadIdx.x;
    if (i >= (size_t)B_ * CN_) return;
    lowlast[i] = XL[i * L_ + (L_ - 1)] + seqfin[i * L_ + (L_ - 1)];
}

__global__ void k_final(const float* __restrict__ lowlast, const float* __restrict__ XH,
                        const float* __restrict__ yfin, const float* __restrict__ alpha,
                        float* __restrict__ out) {
    size_t i = (size_t)blockIdx.x * 256 + threadIdx.x;
    if (i >= (size_t)B_ * CN_) return;
    float a = 1.f / (1.f + __expf(-alpha[0]));
    out[i] = a * lowlast[i] + (1.f - a) * (XH[i * L_ + (L_ - 1)] + yfin[i * LH_ + (LH_ - 1)]);
}

// ---- one temporal-attention layer ------------------------------------------
template<int T>
static void run_layer(const float* seq_in, float* seq_out,
                      const float* c1, const float* c2, const float* wmat,
                      const float* bmat, const float* vmat,
                      const float* g, const float* be,
                      float* f1b, float* f2b, float* f1wb, float* wpad, float* cpad,
                      hipStream_t s) {
    k_wcopy<<<dim3((KPAD_ * C_ + 255) / 256), dim3(256), 0, s>>>(wmat, wpad);
    k_f1<T><<<dim3((N_ * T + 255) / 256, B_), dim3(256), 0, s>>>(seq_in, c1, f1b);
    k_f2<T><<<dim3(C_, B_), dim3(32 * T), 0, s>>>(seq_in, c2, f2b);
    k_f1w<T><<<dim3((B_ * T) / 16, C_ / 16), dim3(32), 0, s>>>(f1b, wpad, f1wb);
    k_attn<T><<<dim3(1), dim3(512), 0, s>>>(f1wb, f2b, bmat, vmat, g, be, cpad);
    k_mix<T><<<dim3(CN_ / 128, B_), dim3(256), 0, s>>>(seq_in, cpad, seq_out);
}

extern "C" void kernel_launch(void* const* d_in, const int* in_sizes, int n_in,
                              void* d_out, int out_size, void* d_ws, size_t ws_size,
                              hipStream_t stream) {
    (void)in_sizes; (void)n_in; (void)out_size; (void)ws_size;
    const float* XL  = (const float*)d_in[0];
    const float* XH  = (const float*)d_in[1];
    const float* lc1 = (const float*)d_in[2];
    const float* lc2 = (const float*)d_in[3];
    const float* lw  = (const float*)d_in[4];
    const float* lb  = (const float*)d_in[5];
    const float* lv  = (const float*)d_in[6];
    const float* lg  = (const float*)d_in[7];
    const float* lbe = (const float*)d_in[8];
    const float* hc1 = (const float*)d_in[9];
    const float* hc2 = (const float*)d_in[10];
    const float* hw  = (const float*)d_in[11];
    const float* hb  = (const float*)d_in[12];
    const float* hv  = (const float*)d_in[13];
    const float* hg  = (const float*)d_in[14];
    const float* hbe = (const float*)d_in[15];
    const float* alpha = (const float*)d_in[16];
    float* out = (float*)d_out;

    // workspace carve (~137 MB): 2 ping-pong seq buffers (T<=13) + smalls
    char* p = (char*)d_ws;
    auto take = [&p](size_t nflt) {
        float* r = (float*)p;
        p += ((nflt * sizeof(float) + 255) / 256) * 256;
        return r;
    };
    const size_t SEQ = (size_t)B_ * CN_ * LH_;
    float* bufA    = take(SEQ);
    float* bufB    = take(SEQ);
    float* lowlast = take((size_t)B_ * CN_);
    float* f1b     = take((size_t)B_ * LH_ * KPAD_);
    float* f2b     = take((size_t)B_ * LH_ * C_);
    float* f1wb    = take((size_t)B_ * LH_ * C_);
    float* wpad    = take((size_t)KPAD_ * C_);
    float* cpad    = take((size_t)B_ * 256);

    const int gridE = (int)(((size_t)B_ * CN_ + 255) / 256);

    // ---- low channel: XL -> A -> B -> A -> B -------------------------------
    const float* cur = XL;
    float* nxt = bufA;
    for (int i = 0; i < 4; ++i) {
        run_layer<L_>(cur, nxt,
                      lc1 + (size_t)i * C_, lc2 + (size_t)i * N_,
                      lw + (size_t)i * N_ * C_,
                      lb + (size_t)i * L_ * L_, lv + (size_t)i * L_ * L_,
                      lg + (size_t)i * L_, lbe + (size_t)i * L_,
                      f1b, f2b, f1wb, wpad, cpad, stream);
        cur = nxt;
        nxt = (nxt == bufA) ? bufB : bufA;
    }
    k_reslast<<<dim3(gridE), dim3(256), 0, stream>>>(XL, cur, lowlast);

    // ---- high channel: pad(XH) -> A -> B -> A -> B -> A --------------------
    const int gridP = (int)((SEQ + 255) / 256);
    k_pad<<<dim3(gridP), dim3(256), 0, stream>>>(XH, bufA);
    const float* hcur = bufA;
    float* hnxt = bufB;
    for (int i = 0; i < 4; ++i) {
        run_layer<LH_>(hcur, hnxt,
                       hc1 + (size_t)i * C_, hc2 + (size_t)i * N_,
                       hw + (size_t)i * N_ * C_,
                       hb + (size_t)i * LH_ * LH_, hv + (size_t)i * LH_ * LH_,
                       hg + (size_t)i * LH_, hbe + (size_t)i * LH_,
                       f1b, f2b, f1wb, wpad, cpad, stream);
        hcur = hnxt;
        hnxt = (hnxt == bufA) ? bufB : bufA;
    }

    // ---- blend -------------------------------------------------------------
    k_final<<<dim3(gridE), dim3(256), 0, stream>>>(lowlast, XH, hcur, alpha, out);
}